// GNN_28948079575208
// MI455X (gfx1250) — compile-verified
//
#include <hip/hip_runtime.h>
#include <hip/hip_bf16.h>

typedef float v2f __attribute__((ext_vector_type(2)));
typedef float v8f __attribute__((ext_vector_type(8)));

// ---------------------------------------------------------------------------
// degree histogram: deg[dst[e]] += 1  (real edges only; self-loop added later)
// ---------------------------------------------------------------------------
__global__ void deg_kernel(const int* __restrict__ dst, float* __restrict__ deg, int E) {
    int i = blockIdx.x * blockDim.x + threadIdx.x;
    int stride = gridDim.x * blockDim.x;
    for (int e = i; e < E; e += stride)
        atomicAdd(&deg[dst[e]], 1.0f);
}

// rs[i] = rsqrt(max(deg[i] + 1, 1))   (the +1 is the self loop)
__global__ void rsqrt_kernel(float* __restrict__ deg, int N) {
    int i = blockIdx.x * blockDim.x + threadIdx.x;
    int stride = gridDim.x * blockDim.x;
    for (int n = i; n < N; n += stride) {
        float d = fmaxf(deg[n] + 1.0f, 1.0f);
        deg[n] = rsqrtf(d);
    }
}

// ---------------------------------------------------------------------------
// C[M x 64] = A[M x K] @ B[K x 64]  via V_WMMA_F32_16X16X4_F32 (full fp32).
// One wave per 16x16 C tile. M must be a multiple of 16 (100000 = 6250*16).
//
// A 16x4 layout : lane&15 = row M, (lane>>4)*2 = K pair -> v2f {K, K+1}
// B 4x16 layout : VGPR i, lanes 0-15 = row i, lanes 16-31 = row i+2
// C 16x16 layout: VGPR i, lanes 0-15 = row i, lanes 16-31 = row i+8
// ---------------------------------------------------------------------------
__global__ void wmma_gemm_kernel(const float* __restrict__ A, const float* __restrict__ B,
                                 float* __restrict__ C, int M, int K) {
    const int lane = threadIdx.x & 31;
    const int waveInBlock = threadIdx.x >> 5;
    const int wavesPerBlock = blockDim.x >> 5;
    const int numTiles = (M >> 4) * 4;                // 64/16 = 4 column tiles
    const int tile = blockIdx.x * wavesPerBlock + waveInBlock;
    if (tile >= numTiles) return;                     // wave-uniform: EXEC stays all-1s
    const int colTile = tile & 3;
    const int rowTile = tile >> 2;
    const int mrow = rowTile * 16 + (lane & 15);      // A row this lane feeds
    const int ncol = colTile * 16 + (lane & 15);      // B/C column this lane holds
    const int kb   = (lane >> 4) << 1;                // 0 or 2

    const float* __restrict__ arow = A + (size_t)mrow * K;
    v8f c = {};
    for (int k0 = 0; k0 < K; k0 += 4) {
        v2f a = *(const v2f*)(arow + k0 + kb);        // {A[m][k0+kb], A[m][k0+kb+1]}
        v2f b;
        b.x = B[(size_t)(k0 + kb)     * 64 + ncol];
        b.y = B[(size_t)(k0 + kb + 1) * 64 + ncol];
        c = __builtin_amdgcn_wmma_f32_16x16x4_f32(false, a, false, b,
                                                  (short)0, c, false, false);
    }

    const int rbase = rowTile * 16 + ((lane >> 4) << 3);   // rows +0 or +8
    float* __restrict__ crow = C + (size_t)rbase * 64 + ncol;
#pragma unroll
    for (int i = 0; i < 8; ++i)
        crow[(size_t)i * 64] = c[i];
}

// ---------------------------------------------------------------------------
// gather -> scale by rs[s]*rs[d] -> scatter-add. One wave per edge.
// 64 features = 2 f32 per lane (wave32). Edges [E, E+N) are self-loops.
// ---------------------------------------------------------------------------
__global__ void scatter_kernel(const float* __restrict__ h, const int* __restrict__ src,
                               const int* __restrict__ dst, const float* __restrict__ rs,
                               float* __restrict__ out, int E, int N) {
    const int lane = threadIdx.x & 31;
    const int wave = (blockIdx.x * blockDim.x + threadIdx.x) >> 5;
    const int nWaves = (gridDim.x * blockDim.x) >> 5;
    const int total = E + N;
    for (int e = wave; e < total; e += nWaves) {
        int s, d;
        float w;
        if (e < E) {
            s = src[e];
            d = dst[e];
            w = rs[s] * rs[d];
        } else {
            s = d = e - E;
            float r = rs[s];
            w = r * r;                        // rsqrt(deg*deg) = 1/deg
        }
        const float* __restrict__ hrow = h + (size_t)s * 64;
        float v0 = hrow[lane]      * w;
        float v1 = hrow[lane + 32] * w;
        float* __restrict__ orow = out + (size_t)d * 64;
        atomicAdd(&orow[lane],      v0);
        atomicAdd(&orow[lane + 32], v1);
    }
}

// h[t] = relu(agg[t] + b[t&63] + p[t])
__global__ void bias_perturb_relu_kernel(const float* __restrict__ agg,
                                         const float* __restrict__ b,
                                         const float* __restrict__ p,
                                         float* __restrict__ h, int total) {
    int i = blockIdx.x * blockDim.x + threadIdx.x;
    int stride = gridDim.x * blockDim.x;
    for (int t = i; t < total; t += stride)
        h[t] = fmaxf(agg[t] + b[t & 63] + p[t], 0.0f);
}

// out[t] += b[t&63] + p[t]   (out already holds the layer-2 aggregation)
__global__ void bias_perturb_kernel(const float* __restrict__ b,
                                    const float* __restrict__ p,
                                    float* __restrict__ out, int total) {
    int i = blockIdx.x * blockDim.x + threadIdx.x;
    int stride = gridDim.x * blockDim.x;
    for (int t = i; t < total; t += stride)
        out[t] = out[t] + b[t & 63] + p[t];
}

// ---------------------------------------------------------------------------
extern "C" void kernel_launch(void* const* d_in, const int* in_sizes, int n_in,
                              void* d_out, int out_size, void* d_ws, size_t ws_size,
                              hipStream_t stream) {
    const float* x  = (const float*)d_in[0];
    const int*   ei = (const int*)  d_in[1];
    const float* W1 = (const float*)d_in[2];
    const float* b1 = (const float*)d_in[3];
    const float* W2 = (const float*)d_in[4];
    const float* b2 = (const float*)d_in[5];
    const float* p1 = (const float*)d_in[6];
    const float* p2 = (const float*)d_in[7];
    float* out = (float*)d_out;

    const int N = in_sizes[0] / 128;       // 100000
    const int E = in_sizes[1] / 2;         // 1600000
    const int* src = ei;                   // edge_index[0]
    const int* dst = ei + E;               // edge_index[1]

    // workspace: rs[N] | bufA[N*64] | bufB[N*64]
    char* ws = (char*)d_ws;
    float* rs   = (float*)ws;
    size_t rsBytes = (((size_t)N * 4) + 255) & ~(size_t)255;
    float* bufA = (float*)(ws + rsBytes);
    float* bufB = bufA + (size_t)N * 64;

    const int totalND = N * 64;
    const int pwBlocks = (totalND + 255) / 256;

    // ---- norm factors ----
    hipMemsetAsync(rs, 0, (size_t)N * 4, stream);
    deg_kernel<<<2048, 256, 0, stream>>>(dst, rs, E);
    rsqrt_kernel<<<(N + 255) / 256, 256, 0, stream>>>(rs, N);

    // ---- layer 1 ----
    {   // bufA = x @ W1
        int numTiles = (N / 16) * 4;
        wmma_gemm_kernel<<<(numTiles + 7) / 8, 256, 0, stream>>>(x, W1, bufA, N, 128);
    }
    hipMemsetAsync(bufB, 0, (size_t)totalND * 4, stream);
    {   // bufB = scatter-add of normalized messages
        int totalEdges = E + N;
        int blocks = (totalEdges + 7) / 8;          // 8 waves per 256-thread block
        scatter_kernel<<<blocks, 256, 0, stream>>>(bufA, src, dst, rs, bufB, E, N);
    }
    // bufA = relu(bufB + b1 + perturb1)
    bias_perturb_relu_kernel<<<pwBlocks, 256, 0, stream>>>(bufB, b1, p1, bufA, totalND);

    // ---- layer 2 ----
    {   // bufB = bufA @ W2 (every element overwritten; no zeroing needed)
        int numTiles = (N / 16) * 4;
        wmma_gemm_kernel<<<(numTiles + 7) / 8, 256, 0, stream>>>(bufA, W2, bufB, N, 64);
    }
    hipMemsetAsync(out, 0, (size_t)out_size * 4, stream);
    {
        int totalEdges = E + N;
        int blocks = (totalEdges + 7) / 8;
        scatter_kernel<<<blocks, 256, 0, stream>>>(bufB, src, dst, rs, out, E, N);
    }
    // out += b2 + perturb2
    bias_perturb_kernel<<<pwBlocks, 256, 0, stream>>>(b2, p2, out, totalND);
}